// Post_Prob_GS_24343874634098
// MI455X (gfx1250) — compile-verified
//
#include <hip/hip_runtime.h>
#include <math.h>

// Post_Prob Gaussian-splat posterior for MI455X (gfx1250, wave32).
//
// z[p,g] = coeff[p,0:8] . feat[0:8,g]   (K=8 GEMM, f32 precision required:
//          gx^2 up to 2.6e5 overflows f16; bf16 mantissa too short)
// -> two V_WMMA_F32_16X16X4_F32 per 16x16 tile.
//
// Bandwidth roofline: 67 MB output / 23.3 TB/s ~= 2.9 us. Design = fully
// fused: z tile lives in LDS (512x16 per wave, ~37 KB), output written once.

typedef __attribute__((ext_vector_type(2))) float v2f;
typedef __attribute__((ext_vector_type(8))) float v8f;

#define BN 8
#define NN 512
#define GN 4096   // 64x64 grid cells

// ---------------------------------------------------------------------------
// Kernel 0: per-point polynomial coefficients of the quadratic form + logdet
// z = c0*gx^2 + c1*gx + c2*gy^2 + c3*gy + c4*gx*gy + c5
// ---------------------------------------------------------------------------
__global__ void __launch_bounds__(256)
coeff_kernel(const float* __restrict__ scale,     // [B,N,2]
             const float* __restrict__ rotation,  // [B,N,1]
             const float* __restrict__ points,    // [B,N,2]
             float* __restrict__ coeff,           // [P,8]
             float* __restrict__ logdet)          // [P]
{
    int p = blockIdx.x * blockDim.x + threadIdx.x;
    if (p >= BN * NN) return;

    // s = clip(16*scale, STRIDE/POST_MIN=4, POST_MAX=64); eigenvalues s^2
    float sx = fminf(fmaxf(16.0f * scale[2 * p + 0], 4.0f), 64.0f);
    float sy = fminf(fmaxf(16.0f * scale[2 * p + 1], 4.0f), 64.0f);
    float s2x = sx * sx, s2y = sy * sy;

    float th = rotation[p];
    float c = cosf(th), si = sinf(th);
    float a  = c * c * s2x + si * si * s2y;   // C00
    float bb = c * si * (s2x - s2y);          // C01
    float d  = si * si * s2x + c * c * s2y;   // C11
    float det = a * d - bb * bb;
    float inv = 1.0f / det;
    float ia = d * inv, ib = -bb * inv, idd = a * inv;  // C^-1

    float px = points[2 * p + 0], py = points[2 * p + 1];
    float* co = coeff + (size_t)p * 8;
    co[0] = -0.5f * ia;                                          // * gx^2
    co[1] = ia * px + ib * py;                                   // * gx
    co[2] = -0.5f * idd;                                         // * gy^2
    co[3] = idd * py + ib * px;                                  // * gy
    co[4] = -ib;                                                 // * gx*gy
    co[5] = -0.5f * (ia * px * px + 2.0f * ib * px * py + idd * py * py);
    co[6] = 0.0f;
    co[7] = 0.0f;
    logdet[p] = logf(det);
}

// ---------------------------------------------------------------------------
// Kernel 1: one wave per (batch, 16-grid-column tile).
// 32 m-tiles x 2x V_WMMA_F32_16X16X4_F32 -> z[512][16] in LDS,
// then fused background row + softmax over 513 rows, single output write.
// ---------------------------------------------------------------------------
__global__ void __launch_bounds__(32)
postprob_kernel(const float* __restrict__ coeff,   // [P,8]
                const float* __restrict__ logdet,  // [P]
                float* __restrict__ out)           // [B,513,G]
{
    __shared__ float zbuf[NN * 17];   // 512 rows x 16 cols, +1 pad vs bank conflicts
    __shared__ float ld_sh[NN];
    __shared__ float redf[32];
    __shared__ int   redi[32];
    __shared__ float redm[32];
    __shared__ float reds[32];

    const int lane = threadIdx.x;          // 0..31, full wave32
    const int b    = blockIdx.x >> 8;      // 256 column tiles per batch
    const int g0   = (blockIdx.x & 255) << 4;
    const int col  = lane & 15;
    const int half = lane >> 4;

    // stage this batch's logdet in LDS
    for (int i = lane; i < NN; i += 32) ld_sh[i] = logdet[b * NN + i];

    // Feature B-fragments for this tile's 16 grid cells (loop invariant).
    // B 4x16 f32 layout: VGPR0 = K(0|2), VGPR1 = K(1|3); lanes 0-15 / 16-31.
    int   g  = g0 + col;
    float gx = (float)((g & 63) * 8 + 4);
    float gy = (float)((g >> 6) * 8 + 4);
    float f0 = gx * gx, f1 = gx, f2 = gy * gy, f3 = gy, f4 = gx * gy, f5 = 1.0f;
    v2f bf0, bf1;                          // K-chunks [0..3] and [4..7]
    bf0.x = half ? f2 : f0;  bf0.y = half ? f3 : f1;
    bf1.x = half ? 0.0f : f4;  bf1.y = half ? 0.0f : f5;

    __syncthreads();

    // A 16x4 f32 layout: lanes 0-15 -> K=0,1 ; lanes 16-31 -> K=2,3.
    const float* Abase = coeff + (size_t)(b * NN) * 8 + half * 2;
    for (int t = 0; t < 32; ++t) {
        const float* ap = Abase + (size_t)(t * 16 + col) * 8;
        v2f a0; a0.x = ap[0]; a0.y = ap[1];   // K-chunk 0..3
        v2f a1; a1.x = ap[4]; a1.y = ap[5];   // K-chunk 4..7
        v8f acc = {};
        acc = __builtin_amdgcn_wmma_f32_16x16x4_f32(
            false, a0, false, bf0, (short)0, acc, false, false);
        acc = __builtin_amdgcn_wmma_f32_16x16x4_f32(
            false, a1, false, bf1, (short)0, acc, false, false);
        // C/D layout: VGPR i -> row i (lanes 0-15) / row 8+i (lanes 16-31)
        int rbase = t * 16 + half * 8;
#pragma unroll
        for (int i = 0; i < 8; ++i)
            zbuf[(rbase + i) * 17 + col] = acc[i];
    }
    __syncthreads();

    // ---- fused background + softmax: each lane owns half a column ----
    const float dthr = -32.0f;             // -0.5*CUT_OFF^2
    const int   r0   = half * 256;
    float zmax = -INFINITY, lmax = -INFINITY;
    int   idx  = r0;
    for (int r = r0; r < r0 + 256; ++r) {
        float z = zbuf[r * 17 + col];
        if (z > zmax) { zmax = z; idx = r; }             // first-occurrence argmax
        lmax = fmaxf(lmax, z - 0.5f * ld_sh[r]);
    }
    redf[lane] = zmax; redi[lane] = idx; redm[lane] = lmax;
    __syncthreads();
    {
        float oz = redf[lane ^ 16]; int oi = redi[lane ^ 16];
        if (oz > zmax || (oz == zmax && oi < idx)) { zmax = oz; idx = oi; }
        lmax = fmaxf(lmax, redm[lane ^ 16]);
    }
    float bk = dthr - zmax - ld_sh[idx];   // BG_RATIO == 1 -> log(det_sel)
    lmax = fmaxf(lmax, bk);

    float ssum = 0.0f;
    for (int r = r0; r < r0 + 256; ++r)
        ssum += __expf(zbuf[r * 17 + col] - 0.5f * ld_sh[r] - lmax);
    reds[lane] = ssum;
    __syncthreads();
    float total = ssum + reds[lane ^ 16] + __expf(bk - lmax);
    float rinv  = 1.0f / total;

    // ---- single coalesced output write: [B, 513, G] ----
    float* outb = out + (size_t)b * 513 * GN + g0 + col;
    for (int r = r0; r < r0 + 256; ++r)
        outb[(size_t)r * GN] =
            __expf(zbuf[r * 17 + col] - 0.5f * ld_sh[r] - lmax) * rinv;
    if (half == 0)
        outb[(size_t)512 * GN] = __expf(bk - lmax) * rinv;
}

// ---------------------------------------------------------------------------
extern "C" void kernel_launch(void* const* d_in, const int* in_sizes, int n_in,
                              void* d_out, int out_size, void* d_ws, size_t ws_size,
                              hipStream_t stream) {
    const float* scale    = (const float*)d_in[0];  // [B,N,2]
    const float* rotation = (const float*)d_in[1];  // [B,N,1]
    const float* points   = (const float*)d_in[2];  // [B,N,2]
    // d_in[3] (st_sizes) is unused by the reference computation.

    float* coeff  = (float*)d_ws;            // 4096*8 floats = 128 KB
    float* logdet = coeff + BN * NN * 8;     // 4096 floats   =  16 KB
    float* out    = (float*)d_out;

    coeff_kernel<<<dim3((BN * NN + 255) / 256), dim3(256), 0, stream>>>(
        scale, rotation, points, coeff, logdet);
    postprob_kernel<<<dim3(BN * (GN / 16)), dim3(32), 0, stream>>>(
        coeff, logdet, out);
}